// social_stgcn_55224689492783
// MI455X (gfx1250) — compile-verified
//
#include <hip/hip_runtime.h>

typedef float v2f __attribute__((ext_vector_type(2)));
typedef float v8f __attribute__((ext_vector_type(8)));

// ---------------- zero fill ----------------
__global__ void k_zero(float* __restrict__ p, long n) {
  long i = (long)blockIdx.x * blockDim.x + threadIdx.x;
  long stride = (long)gridDim.x * blockDim.x;
  for (; i < n; i += stride) p[i] = 0.0f;
}

// ---------------- in-degree histogram ----------------
__global__ void k_degree(const int* __restrict__ ei, int E, unsigned* __restrict__ deg) {
  int e = blockIdx.x * blockDim.x + threadIdx.x;
  if (e < E) atomicAdd(&deg[ei[E + e]], 1u);
}

// ---------------- node prep: dinv factors + scaled GCN features ----------------
__global__ void k_node_prep(const float* __restrict__ x, const float* __restrict__ Wg,
                            const unsigned* __restrict__ deg,
                            float* __restrict__ dinvg, float* __restrict__ dinv0,
                            float* __restrict__ hs, int N) {
  int i = blockIdx.x * blockDim.x + threadIdx.x;
  if (i >= N) return;
  float d  = (float)deg[i];
  float dg = rsqrtf(d + 2.0f);                 // GCN improved=True: A+2I
  float d0 = d > 0.f ? rsqrtf(d) : 0.f;        // Laplacian norm (no self loop)
  dinvg[i] = dg; dinv0[i] = d0;
  float x0 = x[2 * i], x1v = x[2 * i + 1];
  float* o = hs + (size_t)i * 8;
#pragma unroll
  for (int j = 0; j < 5; ++j) o[j] = dg * (x0 * Wg[j] + x1v * Wg[5 + j]);
  o[5] = 0.f; o[6] = 0.f; o[7] = 0.f;
}

// ---------------- edge scatter: agg[dst] += feat[src] (5 comps, stride 8) ----------------
__global__ void k_edge_scatter(const int* __restrict__ ei, int E,
                               const float* __restrict__ feat, float* __restrict__ agg) {
  int e = blockIdx.x * blockDim.x + threadIdx.x;
  if (e >= E) return;
  int s = ei[e], d = ei[E + e];
  const float* f = feat + (size_t)s * 8;
  float4 v = *(const float4*)f;       // 32B-aligned rows -> global_load_b128
  float v4 = f[4];
  float* a = agg + (size_t)d * 8;
  unsafeAtomicAdd(a + 0, v.x);
  unsafeAtomicAdd(a + 1, v.y);
  unsafeAtomicAdd(a + 2, v.z);
  unsafeAtomicAdd(a + 3, v.w);
  unsafeAtomicAdd(a + 4, v4);
}

// ---------------- node pass 2: finish GCN, relu, prep Laplacian features ----------------
// bufA: hs in, x1 out (in place). bufB: agg1 in, reset to 0 (becomes agg2). xs: dinv0*x1.
__global__ void k_node2(const float* __restrict__ bgcn,
                        const float* __restrict__ dinvg, const float* __restrict__ dinv0,
                        float* __restrict__ bufA, float* __restrict__ bufB,
                        float* __restrict__ xs, int N) {
  int i = blockIdx.x * blockDim.x + threadIdx.x;
  if (i >= N) return;
  float dg = dinvg[i], d0 = dinv0[i];
  float* A = bufA + (size_t)i * 8;
  float* B = bufB + (size_t)i * 8;
  float* X = xs   + (size_t)i * 8;
#pragma unroll
  for (int j = 0; j < 5; ++j) {
    float s = B[j]; B[j] = 0.f;                       // recycle agg1 -> agg2
    float hsv = A[j];                                  // hs = dg*h
    float v = dg * s + 2.f * dg * hsv + bgcn[j];       // dg*sum + 2*dg^2*h + b
    float x1 = fmaxf(v, 0.f);
    A[j] = x1;
    X[j] = d0 * x1;
  }
  X[5] = 0.f; X[6] = 0.f; X[7] = 0.f;
}

// ---------------- final: WMMA dense transform + gates + output linear ----------------
// One wave (32 threads) per block; 16 nodes per tile; 3x V_WMMA_F32_16X16X4_F32.
__global__ void __launch_bounds__(32)
k_final(const float* __restrict__ x1buf, const float* __restrict__ aggbuf,
        const float* __restrict__ dinv0,
        const float* __restrict__ Wxz, const float* __restrict__ Wxh,
        const float* __restrict__ bxz, const float* __restrict__ bhz,
        const float* __restrict__ bxh, const float* __restrict__ bhh,
        const float* __restrict__ Wlin, const float* __restrict__ blin,
        float* __restrict__ out, int N, int ntiles) {
  __shared__ float tile[16][12];
  const int l = threadIdx.x;          // 0..31
  const int m = l & 15;               // A row / B,D column
  const int hi = l >> 4;              // lane half
  const int koff = hi * 2;            // K offset for this half (f32 16x16x4 layout)
  const int n = m;

  // B fragments (wave-invariant): Wc[12][16] = [Wxz0;Wxz1|0] cols 0..4, [Wxh0;Wxh1|0] cols 5..9
  v2f bfrag[3];
#pragma unroll
  for (int t = 0; t < 3; ++t) {
#pragma unroll
    for (int dd = 0; dd < 2; ++dd) {
      int k = 4 * t + koff + dd;
      float w = 0.f;
      if (n < 5) {
        if (k < 5)       w = Wxz[k * 5 + n];
        else if (k < 10) w = Wxz[25 + (k - 5) * 5 + n];
      } else if (n < 10) {
        int nn = n - 5;
        if (k < 5)       w = Wxh[k * 5 + nn];
        else if (k < 10) w = Wxh[25 + (k - 5) * 5 + nn];
      }
      bfrag[t][dd] = w;
    }
  }

  // hoist epilogue constants (wave-uniform scalar loads, once)
  float zbias[5], hbias[5], wl[15], bl[3];
#pragma unroll
  for (int j = 0; j < 5; ++j) { zbias[j] = bxz[j] + bhz[j]; hbias[j] = bxh[j] + bhh[j]; }
#pragma unroll
  for (int j = 0; j < 15; ++j) wl[j] = Wlin[j];
#pragma unroll
  for (int o = 0; o < 3; ++o) bl[o] = blin[o];

  for (int tb = blockIdx.x; tb < ntiles; tb += gridDim.x) {
    int node = tb * 16 + m;
    bool valid = node < N;

    // Uniform vector loads of the node's x1 row and aggregate row, then lane-half selects.
    float d0 = 0.f, x4 = 0.f, g4 = 0.f;
    float4 xv = make_float4(0.f, 0.f, 0.f, 0.f);
    float4 gv = make_float4(0.f, 0.f, 0.f, 0.f);
    if (valid) {
      d0 = dinv0[node];
      const float* xr = x1buf + (size_t)node * 8;
      xv = *(const float4*)xr;  x4 = xr[4];
      const float* gr = aggbuf + (size_t)node * 8;
      gv = *(const float4*)gr;  g4 = gr[4];
    }
    float lx0 = -d0 * gv.x, lx1 = -d0 * gv.y, lx2 = -d0 * gv.z,
          lx3 = -d0 * gv.w, lx4 = -d0 * g4;

    // A fragments: feat k<5 -> x1[k]; k<10 -> lX[k-5]; else 0. K tile layout: lanes16-31 = K+2.
    v2f a0, a1, a2;
    a0[0] = hi ? xv.z : xv.x;   a0[1] = hi ? xv.w : xv.y;   // k = {0,1} / {2,3}
    a1[0] = hi ? lx1  : x4;     a1[1] = hi ? lx2  : lx0;    // k = {4,5} / {6,7}
    a2[0] = hi ? 0.f  : lx3;    a2[1] = hi ? 0.f  : lx4;    // k = {8,9} / {10,11}

    v8f c = {};
    c = __builtin_amdgcn_wmma_f32_16x16x4_f32(false, a0, false, bfrag[0], (short)0, c, false, false);
    c = __builtin_amdgcn_wmma_f32_16x16x4_f32(false, a1, false, bfrag[1], (short)0, c, false, false);
    c = __builtin_amdgcn_wmma_f32_16x16x4_f32(false, a2, false, bfrag[2], (short)0, c, false, false);

    if (n < 10) {
#pragma unroll
      for (int r = 0; r < 8; ++r) tile[r + hi * 8][n] = c[r];  // D: vgpr r -> row r(+8)
    }
    __syncthreads();
    if (l < 16) {
      int nd = tb * 16 + l;
      if (nd < N) {
        float hrel[5];
#pragma unroll
        for (int j = 0; j < 5; ++j) {
          float zp = tile[l][j]     + zbias[j];
          float hp = tile[l][j + 5] + hbias[j];
          float Z  = 1.f / (1.f + __expf(-zp));
          float Ht = tanhf(hp);
          hrel[j] = fmaxf((1.f - Z) * Ht, 0.f);   // H = (1-Z)*H~ (H0==0), then ReLU
        }
#pragma unroll
        for (int o = 0; o < 3; ++o) {
          float s = bl[o];
#pragma unroll
          for (int j = 0; j < 5; ++j) s += hrel[j] * wl[j * 3 + o];
          out[(size_t)nd * 3 + o] = s;
        }
      }
    }
    __syncthreads();
  }
}

extern "C" void kernel_launch(void* const* d_in, const int* in_sizes, int n_in,
                              void* d_out, int out_size, void* d_ws, size_t ws_size,
                              hipStream_t stream) {
  const float* x    = (const float*)d_in[0];
  const int*   ei   = (const int*)d_in[1];
  const float* Wg   = (const float*)d_in[2];
  const float* bg   = (const float*)d_in[3];
  const float* Wlin = (const float*)d_in[4];
  const float* blin = (const float*)d_in[5];
  const float* Wxz  = (const float*)d_in[6];
  const float* bxz  = (const float*)d_in[7];
  const float* bhz  = (const float*)d_in[9];
  const float* Wxh  = (const float*)d_in[14];
  const float* bxh  = (const float*)d_in[15];
  const float* bhh  = (const float*)d_in[17];

  const int N = in_sizes[0] / 2;
  const int E = in_sizes[1] / 2;
  const size_t Np = ((size_t)N + 3) & ~(size_t)3;   // keep 16B alignment of rows

  float* W = (float*)d_ws;
  unsigned* deg = (unsigned*)W;           // Np u32
  float* dinvg  = W + Np;                 // Np f32
  float* dinv0  = W + 2 * Np;             // Np f32
  float* bufA   = W + 3 * Np;             // 8*Np : hs -> x1
  float* bufB   = W + 11 * Np;            // 8*Np : agg1 -> agg2
  float* xs     = W + 19 * Np;            // 8*Np : dinv0*x1

  const int tb = 256;
  const int nbN = (N + tb - 1) / tb;
  const int nbE = (E + tb - 1) / tb;

  k_zero<<<1024, tb, 0, stream>>>((float*)deg, (long)Np);
  k_zero<<<2048, tb, 0, stream>>>(bufB, (long)Np * 8);
  k_degree<<<nbE, tb, 0, stream>>>(ei, E, deg);
  k_node_prep<<<nbN, tb, 0, stream>>>(x, Wg, deg, dinvg, dinv0, bufA, N);
  k_edge_scatter<<<nbE, tb, 0, stream>>>(ei, E, bufA, bufB);          // GCN aggregate
  k_node2<<<nbN, tb, 0, stream>>>(bg, dinvg, dinv0, bufA, bufB, xs, N);
  k_edge_scatter<<<nbE, tb, 0, stream>>>(ei, E, xs, bufB);            // Laplacian aggregate
  const int ntiles = (N + 15) / 16;
  k_final<<<2048, 32, 0, stream>>>(bufA, bufB, dinv0, Wxz, Wxh, bxz, bhz, bxh, bhh,
                                   Wlin, blin, (float*)d_out, N, ntiles);
}